// GINLayer_12764642804257
// MI455X (gfx1250) — compile-verified
//
#include <hip/hip_runtime.h>

#define N_NODES 100000
#define N_EDGES 1000000
#define DIM 64
#define BN_EPS 1e-5f

typedef __attribute__((ext_vector_type(2))) float v2f;
typedef __attribute__((ext_vector_type(8))) float v8f;

// ---------------------------------------------------------------------------
// Kernel 1: h = x (float4 copy), zero the stats accumulator (idempotent).
// ---------------------------------------------------------------------------
__global__ void gin_init_kernel(const float* __restrict__ x,
                                float* __restrict__ h,
                                float* __restrict__ stats) {
    int t = blockIdx.x * blockDim.x + threadIdx.x;
    if (t < 128) stats[t] = 0.0f;
    if (t < N_NODES * DIM / 4) {
        ((float4*)h)[t] = ((const float4*)x)[t];
    }
}

// ---------------------------------------------------------------------------
// Kernel 2: scatter-add over edges: h[dst] += x[src].
// 16 threads per edge, each handles 4 consecutive features (float4 gather,
// 4 scalar f32 atomics). x (25.6MB) is L2-resident on MI455X (192MB L2).
// ---------------------------------------------------------------------------
__global__ void gin_scatter_kernel(const float* __restrict__ x,
                                   const int* __restrict__ edge_index,
                                   float* __restrict__ h) {
    long long t = (long long)blockIdx.x * blockDim.x + threadIdx.x;
    if (t >= (long long)N_EDGES * 16) return;
    int e = (int)(t >> 4);
    int q = ((int)t & 15) * 4;
    int src = edge_index[e];
    int dst = edge_index[N_EDGES + e];
    // defensive clamp (no-op for valid inputs)
    src = src < 0 ? 0 : (src >= N_NODES ? N_NODES - 1 : src);
    dst = dst < 0 ? 0 : (dst >= N_NODES ? N_NODES - 1 : dst);
    float4 v = *(const float4*)(x + (size_t)src * DIM + q);
    float* o = h + (size_t)dst * DIM + q;
    atomicAdd(o + 0, v.x);
    atomicAdd(o + 1, v.y);
    atomicAdd(o + 2, v.z);
    atomicAdd(o + 3, v.w);
}

// ---------------------------------------------------------------------------
// Kernel 3: fused 2-layer MLP via V_WMMA_F32_16X16X4_F32, persistent waves.
//
//  * W1/W2 staged in LDS once per block in "paired-K" layout:
//       Wp[(k>>1)*128 + n*2 + (k&1)] = W[k][n]
//    so a lane's B fragment (W[k][n], W[k+1][n]) is one ds_load_b64.
//  * b1/b2 staged in LDS; single uniform __syncthreads() before divergence.
//  * Each wave32 grid-strides over 16-node tiles; per-tile weight traffic is
//    LDS-only (global weight traffic: 32KB per block instead of per tile).
//  * Per-wave LDS staging buffer between the two GEMMs; same-wave DS
//    ordering + explicit s_wait_dscnt, no further barriers.
//
// Layouts (CDNA5 ISA 7.12.2):
//   A 16x4 f32 : lane&15 = M;  VGPR0/1 = K = khalf, khalf+1 (khalf = lane<16?0:2)
//   B 4x16 f32 : lane&15 = N;  VGPR0/1 = K = khalf, khalf+1
//   C/D 16x16  : elem r -> M = r + (lane<16?0:8), N = lane&15
// ---------------------------------------------------------------------------
__global__ void gin_mlp_wmma_kernel(const float* __restrict__ h,
                                    const float* __restrict__ W1,
                                    const float* __restrict__ b1,
                                    const float* __restrict__ W2,
                                    const float* __restrict__ b2,
                                    float* __restrict__ h2,
                                    float* __restrict__ stats) {
    __shared__ float Wlds[2][DIM * DIM];   // 32 KB: W1, W2 in paired-K layout
    __shared__ float blds[2 * DIM];        // staged b1, b2
    __shared__ float tbuf[4][16 * DIM];    // per-wave inter-GEMM staging

    const int tid  = threadIdx.x;          // 0..127
    const int wave = tid >> 5;
    const int lane = tid & 31;

    // ---- cooperative staging (uniform across block, before any exit) ----
    {
        const float4* w1v = (const float4*)W1;
        const float4* w2v = (const float4*)W2;
#pragma unroll
        for (int i = 0; i < (DIM * DIM / 4) / 128; ++i) {   // 8 iters
            int idx = tid + i * 128;       // float4 index: row k, cols n4..n4+3
            int k   = idx >> 4;            // 16 float4 per 64-col row
            int n4  = (idx & 15) * 4;
            float4 w1q = w1v[idx];
            float4 w2q = w2v[idx];
            float* d1 = Wlds[0] + (k >> 1) * 128 + (k & 1);
            float* d2 = Wlds[1] + (k >> 1) * 128 + (k & 1);
            d1[(n4 + 0) * 2] = w1q.x;  d1[(n4 + 1) * 2] = w1q.y;
            d1[(n4 + 2) * 2] = w1q.z;  d1[(n4 + 3) * 2] = w1q.w;
            d2[(n4 + 0) * 2] = w2q.x;  d2[(n4 + 1) * 2] = w2q.y;
            d2[(n4 + 2) * 2] = w2q.z;  d2[(n4 + 3) * 2] = w2q.w;
        }
        if (tid < DIM) blds[tid] = b1[tid];
        else           blds[tid] = b2[tid - DIM];           // tid in [64,128)
    }
    __syncthreads();

    const int m      = lane & 15;              // M index (A) and N index (B/C)
    const int khalf  = (lane >> 4) << 1;       // 0 or 2
    const int rowoff = (lane >> 4) << 3;       // 0 or 8 (C/D row offset)
    float* my = tbuf[wave];

    // ---- persistent grid-stride loop over 16-node tiles ------------------
    for (int tile = blockIdx.x * 4 + wave; tile * 16 < N_NODES;
         tile += gridDim.x * 4) {
        const int node0 = tile * 16;

        // ---- Layer 1: relu(h @ W1 + b1) ----------------------------------
        v8f c0 = {}, c1 = {}, c2 = {}, c3 = {};
        for (int kb = 0; kb < DIM; kb += 4) {
            const float* ap = h + (size_t)(node0 + m) * DIM + kb + khalf;
            v2f a; a.x = ap[0]; a.y = ap[1];
            const float* bp = Wlds[0] + ((kb + khalf) >> 1) * 128;  // paired-K row
            v2f b;
            b = *(const v2f*)(bp + (m +  0) * 2);
            c0 = __builtin_amdgcn_wmma_f32_16x16x4_f32(false, a, false, b, (short)0, c0, false, false);
            b = *(const v2f*)(bp + (m + 16) * 2);
            c1 = __builtin_amdgcn_wmma_f32_16x16x4_f32(false, a, false, b, (short)0, c1, false, false);
            b = *(const v2f*)(bp + (m + 32) * 2);
            c2 = __builtin_amdgcn_wmma_f32_16x16x4_f32(false, a, false, b, (short)0, c2, false, false);
            b = *(const v2f*)(bp + (m + 48) * 2);
            c3 = __builtin_amdgcn_wmma_f32_16x16x4_f32(false, a, false, b, (short)0, c3, false, false);
        }

        // bias + relu, stash in per-wave LDS row-major [row][feat]
#pragma unroll
        for (int r = 0; r < 8; ++r) {
            int row = r + rowoff;
            my[row * DIM +  0 + m] = fmaxf(c0[r] + blds[ 0 + m], 0.0f);
            my[row * DIM + 16 + m] = fmaxf(c1[r] + blds[16 + m], 0.0f);
            my[row * DIM + 32 + m] = fmaxf(c2[r] + blds[32 + m], 0.0f);
            my[row * DIM + 48 + m] = fmaxf(c3[r] + blds[48 + m], 0.0f);
        }
        asm volatile("s_wait_dscnt 0" ::: "memory");

        // ---- Layer 2: t @ W2 + b2 ----------------------------------------
        v8f d0 = {}, d1 = {}, d2 = {}, d3 = {};
        for (int kb = 0; kb < DIM; kb += 4) {
            v2f a;
            a.x = my[m * DIM + kb + khalf];
            a.y = my[m * DIM + kb + khalf + 1];
            const float* bp = Wlds[1] + ((kb + khalf) >> 1) * 128;  // paired-K row
            v2f b;
            b = *(const v2f*)(bp + (m +  0) * 2);
            d0 = __builtin_amdgcn_wmma_f32_16x16x4_f32(false, a, false, b, (short)0, d0, false, false);
            b = *(const v2f*)(bp + (m + 16) * 2);
            d1 = __builtin_amdgcn_wmma_f32_16x16x4_f32(false, a, false, b, (short)0, d1, false, false);
            b = *(const v2f*)(bp + (m + 32) * 2);
            d2 = __builtin_amdgcn_wmma_f32_16x16x4_f32(false, a, false, b, (short)0, d2, false, false);
            b = *(const v2f*)(bp + (m + 48) * 2);
            d3 = __builtin_amdgcn_wmma_f32_16x16x4_f32(false, a, false, b, (short)0, d3, false, false);
        }

        // pre-BN output -> global + LDS (for the stats reduction)
#pragma unroll
        for (int r = 0; r < 8; ++r) {
            int row = r + rowoff;
            float v0 = d0[r] + blds[DIM +  0 + m];
            float v1 = d1[r] + blds[DIM + 16 + m];
            float v2 = d2[r] + blds[DIM + 32 + m];
            float v3 = d3[r] + blds[DIM + 48 + m];
            float* g = h2 + (size_t)(node0 + row) * DIM;
            g[ 0 + m] = v0;  g[16 + m] = v1;  g[32 + m] = v2;  g[48 + m] = v3;
            my[row * DIM +  0 + m] = v0;
            my[row * DIM + 16 + m] = v1;
            my[row * DIM + 32 + m] = v2;
            my[row * DIM + 48 + m] = v3;
        }
        asm volatile("s_wait_dscnt 0" ::: "memory");

        // per-feature partial sums over this tile's 16 rows (2 features/lane)
#pragma unroll
        for (int fi = 0; fi < 2; ++fi) {
            int f = lane * 2 + fi;
            float s = 0.0f, sq = 0.0f;
#pragma unroll
            for (int row = 0; row < 16; ++row) {
                float v = my[row * DIM + f];
                s += v; sq += v * v;
            }
            atomicAdd(&stats[f], s);
            atomicAdd(&stats[DIM + f], sq);
        }
    }
}

// ---------------------------------------------------------------------------
// Kernel 4: fold batch stats + gamma/beta into per-feature scale & shift.
// ---------------------------------------------------------------------------
__global__ void gin_bnparam_kernel(const float* __restrict__ stats,
                                   const float* __restrict__ gamma,
                                   const float* __restrict__ beta,
                                   float* __restrict__ scsh) {
    int f = threadIdx.x;
    if (f >= DIM) return;
    float inv_n = 1.0f / (float)N_NODES;
    float mean  = stats[f] * inv_n;
    float var   = stats[DIM + f] * inv_n - mean * mean;
    float s     = gamma[f] * rsqrtf(var + BN_EPS);
    scsh[f]       = s;
    scsh[DIM + f] = beta[f] - mean * s;
}

// ---------------------------------------------------------------------------
// Kernel 5: out = relu(h2 * scale + shift), float4-vectorized.
// ---------------------------------------------------------------------------
__global__ void gin_final_kernel(const float* __restrict__ h2,
                                 const float* __restrict__ scsh,
                                 float* __restrict__ out) {
    int t = blockIdx.x * blockDim.x + threadIdx.x;
    if (t >= N_NODES * DIM / 4) return;
    int f = (t * 4) & (DIM - 1);
    float4 v = ((const float4*)h2)[t];
    float4 r;
    r.x = fmaxf(v.x * scsh[f + 0] + scsh[DIM + f + 0], 0.0f);
    r.y = fmaxf(v.y * scsh[f + 1] + scsh[DIM + f + 1], 0.0f);
    r.z = fmaxf(v.z * scsh[f + 2] + scsh[DIM + f + 2], 0.0f);
    r.w = fmaxf(v.w * scsh[f + 3] + scsh[DIM + f + 3], 0.0f);
    ((float4*)out)[t] = r;
}

// ---------------------------------------------------------------------------
extern "C" void kernel_launch(void* const* d_in, const int* in_sizes, int n_in,
                              void* d_out, int out_size, void* d_ws, size_t ws_size,
                              hipStream_t stream) {
    const float* x     = (const float*)d_in[0];
    const int*   ei    = (const int*)  d_in[1];   // edge_index [2, N_EDGES]
    const float* W1    = (const float*)d_in[2];
    const float* b1    = (const float*)d_in[3];
    const float* W2    = (const float*)d_in[4];
    const float* b2    = (const float*)d_in[5];
    const float* gamma = (const float*)d_in[6];
    const float* beta  = (const float*)d_in[7];
    float*       out   = (float*)d_out;

    char*  ws    = (char*)d_ws;
    float* h     = (float*)(ws);                                   // 25.6 MB
    float* h2    = (float*)(ws + (size_t)N_NODES * DIM * 4);       // 25.6 MB
    float* stats = (float*)(ws + (size_t)N_NODES * DIM * 8);       // 512 B
    float* scsh  = stats + 128;                                    // 512 B

    const int nvec4 = N_NODES * DIM / 4;                           // 1.6M

    gin_init_kernel<<<(nvec4 + 255) / 256, 256, 0, stream>>>(x, h, stats);
    gin_scatter_kernel<<<(N_EDGES * 16) / 256, 256, 0, stream>>>(x, ei, h);

    // persistent: 256 blocks * 4 waves = 1024 waves covering 6250 tiles
    gin_mlp_wmma_kernel<<<256, 128, 0, stream>>>(h, W1, b1, W2, b2, h2, stats);

    gin_bnparam_kernel<<<1, 64, 0, stream>>>(stats, gamma, beta, scsh);
    gin_final_kernel<<<(nvec4 + 255) / 256, 256, 0, stream>>>(h2, scsh, out);
}